// GDKR_42958262895149
// MI455X (gfx1250) — compile-verified
//
#include <hip/hip_runtime.h>
#include <math.h>

typedef __attribute__((ext_vector_type(2))) float v2f;
typedef __attribute__((ext_vector_type(8))) float v8f;

#define NF 32
#define NFEAT 1024
#define NT 512
#define NCAT 1536
#define NB 16
#define MEMSZ 16384
#define MOM 0.85f
#define MMLR 0.03f
#define INV_SQRT_NF 0.17677669529663687f
#define TWO_PI 6.283185307179586f

// ---- d_out float offsets (outputs concatenated in reference return order) ----
#define O_MEM  0            // memory            16384*512
#define O_TMEM 8388608      // t_memory (as f32) 16384
#define O_FEAT 8404992      // feat_memory       16384*1024
#define O_ARE  25182208     // amp_re            32*512
#define O_AIM  25198592     // amp_im            32*512
#define O_FRQ  25214976     // freq              32
#define O_RP   25215008     // res_proj          1536*512
#define O_BIAS 26001440     // bias              512
#define O_FM   26001952     // feat_mean         1024
#define O_AGR  26002976     // amp_grad_re       32*512
#define O_AGI  26019360     // amp_grad_im       32*512
#define O_FGM  26035744     // freq_grad         32
#define O_RPG  26035776     // res_proj_grad     1536*512

// ---- workspace float offsets ----
#define W_LR    0
#define W_AN    1
#define W_FN    2
#define W_RN    3
#define W_FG    68      // 32
#define W_ZB    1124    // 16*512 gathered targets
#define W_Z     9316    // 16*512 model output z_
#define W_U     17508   // 16*1536 concat features
#define W_R     42084   // 16*512 r = u @ rp
#define W_G     50276   // 16*512 dTotal/dr
#define W_E     58468   // 16*512 dL2/dz
#define W_DZ    66660   // 16*512 total dTotal/dz
#define W_AGR   74852   // 32*512
#define W_AGI   91236   // 32*512

__device__ __forceinline__ float sgn(float x) {
  return (x > 0.f) ? 1.f : ((x < 0.f) ? -1.f : 0.f);
}

__device__ __forceinline__ v8f wmma4(v2f a, v2f b, v8f c) {
  // V_WMMA_F32_16X16X4_F32 : D = A(16x4) * B(4x16) + C(16x16), fp32
  return __builtin_amdgcn_wmma_f32_16x16x4_f32(false, a, false, b, (short)0, c,
                                               false, false);
}

// ---------------- state copy-out (HBM-bound, b128 path) ----------------
__global__ void k_copy4(const float4* __restrict__ src, float4* __restrict__ dst, int n4) {
  int i = blockIdx.x * blockDim.x + threadIdx.x;
  int stride = gridDim.x * blockDim.x;
  for (; i < n4; i += stride) dst[i] = src[i];
}

__global__ void k_tmem(const int* __restrict__ tmem, const int* p_t, const int* p_ds,
                       float* __restrict__ out) {
  int i = blockIdx.x * blockDim.x + threadIdx.x;
  if (i < MEMSZ) {
    int ds = *p_ds;
    out[O_TMEM + i] = (float)((i == ds) ? *p_t : tmem[i]);
  }
}

__global__ void k_fixrows(const float* __restrict__ zin, const float* __restrict__ xin,
                          const int* p_ds, float* __restrict__ out) {
  int t = blockIdx.x * blockDim.x + threadIdx.x;
  int ds = *p_ds;
  if (t < NT)    out[O_MEM + ds * NT + t] = zin[t];
  if (t < NFEAT) out[O_FEAT + ds * NFEAT + t] = xin[t];
}

// bias / feat_mean running means + lr
__global__ void k_init(const float* __restrict__ bias_in, const float* __restrict__ zin,
                       const float* __restrict__ fm_in, const float* __restrict__ xin,
                       const int* p_ds, float* __restrict__ out, float* __restrict__ ws) {
  int t = threadIdx.x;
  float dsn = (float)(*p_ds + 1);
  if (t < NT)    out[O_BIAS + t] = (bias_in[t] * dsn + zin[t]) / (dsn + 1.f);
  if (t < NFEAT) out[O_FM + t]   = (fm_in[t] * dsn + xin[t]) / (dsn + 1.f);
  if (t == 0)    ws[W_LR] = powf(0.977f, dsn);
}

// =======================================================================
// Fused per-batch optimizer step: one workgroup (32 wave32 on one WGP).
// Phase barriers via s_barrier; 16x16 WMMA tiles map 1 tile/wave (32 tiles)
// for the two forward/backward GEMMs and 96 tiles/wave for the update GEMM.
// =======================================================================
__global__ void __launch_bounds__(1024) k_batch(
    const int* __restrict__ ts, const int* p_t, const int* p_ds,
    const int* __restrict__ tmem, const float* __restrict__ mem,
    const float* __restrict__ zin, const float* __restrict__ xin,
    const float* __restrict__ feat, const float* __restrict__ cfreq,
    float* __restrict__ out, float* __restrict__ ws, int b) {
  __shared__ float sc[NB * NF];   // cos table
  __shared__ float ss[NB * NF];   // sin table
  __shared__ float sdfr[NF];      // dfr/dfq
  __shared__ float stb[NB];       // t_b
  __shared__ float red[1024];     // reduction scratch
  __shared__ int   sb[NB];        // batch indices

  const int t = threadIdx.x;
  const int lane = t & 31, wv = t >> 5;       // 32 waves
  const int half = lane >> 4, q = lane & 15;
  const int ds = *p_ds;

  // ---- phase 0: batch gather + trig tables ----
  if (t < NB) {
    int bi = ts[b * NB + t];
    sb[t] = bi;
    stb[t] = (bi == ds) ? (float)(*p_t) : (float)tmem[bi];
  }
  __syncthreads();
  if (t < NB * NF) {
    int i = t >> 5, k = t & 31;
    float fq = out[O_FRQ + k];                 // current freq (updated per batch)
    float th = tanhf(fq);
    float arg = cfreq[k] + 2.f * th;           // 64/N_FUNC == 2
    float sig = 1.f / (1.f + expf(-arg));
    float ph = TWO_PI * stb[i] * (0.5f * sig);
    sc[t] = cosf(ph);
    ss[t] = sinf(ph);
    if (i == 0) sdfr[k] = sig * (1.f - sig) * (1.f - th * th);
  }
  for (int idx = t; idx < NB * NT; idx += 1024) {
    int i = idx >> 9, j = idx & 511;
    int bi = sb[i];
    ws[W_ZB + idx] = (bi == ds) ? zin[j] : mem[(size_t)bi * NT + j];
  }
  __syncthreads();

  // ---- phase 1: u assembly + z_ evaluation ----
  for (int idx = t; idx < NB * NCAT; idx += 1024) {
    int i = idx / NCAT, f = idx - i * NCAT;
    int bi = sb[i];
    if (f < NFEAT) {
      float xv = (bi == ds) ? xin[f] : feat[(size_t)bi * NFEAT + f];
      ws[W_U + idx] = xv - out[O_FM + f];
    } else {
      int j = f - NFEAT;
      const float* ar = out + O_ARE;
      const float* ai = out + O_AIM;
      float acc = 0.f;
#pragma unroll
      for (int k = 0; k < NF; k++)
        acc += sc[i * NF + k] * ar[k * NT + j] - ss[i * NF + k] * ai[k * NT + j];
      float zc = acc * INV_SQRT_NF;            // z_ - bias
      ws[W_Z + i * NT + j] = zc + out[O_BIAS + j];
      ws[W_U + idx] = zc;
    }
  }
  __threadfence_block();
  __syncthreads();

  // ---- phase 2: r = u(16x1536) @ rp(1536x512), one WMMA tile per wave ----
  {
    int jb = wv * 16;
    const float* u = ws + W_U;
    const float* rp = out + O_RP;
    v8f acc = {0.f, 0.f, 0.f, 0.f, 0.f, 0.f, 0.f, 0.f};
    for (int k = 0; k < NCAT; k += 4) {
      int kk = k + 2 * half;                   // lanes 0-15: K{k,k+1}; 16-31: K{k+2,k+3}
      v2f a, bb;
      a.x = u[q * NCAT + kk];
      a.y = u[q * NCAT + kk + 1];
      bb.x = rp[kk * NT + jb + q];
      bb.y = rp[(kk + 1) * NT + jb + q];
      acc = wmma4(a, bb, acc);
    }
#pragma unroll
    for (int v = 0; v < 8; v++)                // C: VGPR v -> rows v / v+8
      ws[W_R + (v + 8 * half) * NT + jb + q] = acc[v];
  }
  __threadfence_block();
  __syncthreads();

  // ---- phase 3: g = dTotal/dr, e = dL2/dz ----
  for (int idx = t; idx < NB * NT; idx += 1024) {
    float rv = ws[W_R + idx];
    float den = fabsf(rv) + 1.f;
    float zv = ws[W_Z + idx];
    float z2 = zv + rv / den;
    float zb = ws[W_ZB + idx];
    ws[W_G + idx] = (-MMLR / 16.f) * sgn(zb - z2) / (den * den);
    ws[W_E + idx] = (-1.f / 16.f) * sgn(zb - zv);
  }
  __threadfence_block();
  __syncthreads();

  // ---- phase 4: h = g(16x512) @ rp_z^T ; dz = e + h ----
  {
    int jb = wv * 16;
    const float* g = ws + W_G;
    const float* rp = out + O_RP;
    v8f acc = {0.f, 0.f, 0.f, 0.f, 0.f, 0.f, 0.f, 0.f};
    for (int k = 0; k < NT; k += 4) {
      int kk = k + 2 * half;
      v2f a, bb;
      a.x = g[q * NT + kk];
      a.y = g[q * NT + kk + 1];
      bb.x = rp[(NFEAT + jb + q) * NT + kk];   // B[k'=j'][n=j] = rp[1024+j][j']
      bb.y = rp[(NFEAT + jb + q) * NT + kk + 1];
      acc = wmma4(a, bb, acc);
    }
#pragma unroll
    for (int v = 0; v < 8; v++) {
      int o = (v + 8 * half) * NT + jb + q;
      ws[W_DZ + o] = ws[W_E + o] + acc[v];
    }
  }
  __threadfence_block();
  __syncthreads();

  // ---- phase 5: amp gradients (incl. L1 amp-decay term) ----
  for (int idx = t; idx < NF * NT; idx += 1024) {
    int k = idx >> 9, j = idx & 511;
    float sr = 0.f, si = 0.f;
#pragma unroll
    for (int i = 0; i < NB; i++) {
      float d = ws[W_DZ + i * NT + j];
      sr += sc[i * NF + k] * d;
      si -= ss[i * NF + k] * d;
    }
    float ar = out[O_ARE + idx], ai = out[O_AIM + idx];
    float amp2 = ar * ar + ai * ai;
    float p = 0.005f * powf(amp2, -0.75f);     // d(0.01*(amp2)^{1/4})
    ws[W_AGR + idx] = sr * INV_SQRT_NF + p * ar;
    ws[W_AGI + idx] = si * INV_SQRT_NF + p * ai;
  }
  __threadfence_block();
  __syncthreads();

  // ---- phase 6: deterministic reductions (anorm, fg+fnorm, rnorm) ----
  {
    float acc = 0.f;
    for (int idx = t; idx < NF * NT; idx += 1024) {
      float a = ws[W_AGR + idx], bb = ws[W_AGI + idx];
      acc += a * a + bb * bb;
    }
    red[t] = acc;
  }
  __syncthreads();
  for (int s = 512; s > 0; s >>= 1) {
    if (t < s) red[t] += red[t + s];
    __syncthreads();
  }
  if (t == 0) ws[W_AN] = sqrtf(red[0]) + 1.f;
  __syncthreads();
  if (t < NF * NB) {                           // fg: (k,i) partial dots
    int k = t >> 4, i = t & 15;
    float ck = sc[i * NF + k], sk = ss[i * NF + k];
    const float* ar = out + O_ARE + k * NT;
    const float* ai = out + O_AIM + k * NT;
    const float* dz = ws + W_DZ + i * NT;
    float a2 = 0.f;
    for (int j = 0; j < NT; j++) a2 += dz[j] * (-sk * ar[j] - ck * ai[j]);
    red[t] = a2 * stb[i];
  }
  __syncthreads();
  if (t < NF) {
    float s2 = 0.f;
    for (int i = 0; i < 16; i++) s2 += red[t * 16 + i];
    ws[W_FG + t] = s2 * TWO_PI * INV_SQRT_NF * sdfr[t];
  }
  __syncthreads();
  if (t == 0) {
    float s2 = 0.f;
    for (int k = 0; k < NF; k++) { float v = ws[W_FG + k]; s2 += v * v; }
    ws[W_FN] = sqrtf(s2) + 1.f;
  }
  __syncthreads();
  {                                            // rnorm: ||u^T g||_F^2 via Gram matrices
    if (t < 256) {
      int i1 = t >> 4, i2 = t & 15;
      const float* u1 = ws + W_U + i1 * NCAT;
      const float* u2 = ws + W_U + i2 * NCAT;
      float a3 = 0.f;
      for (int f = 0; f < NCAT; f++) a3 += u1[f] * u2[f];
      red[t] = a3;
    } else if (t < 512) {
      int p = t - 256;
      int i1 = p >> 4, i2 = p & 15;
      const float* g1 = ws + W_G + i1 * NT;
      const float* g2 = ws + W_G + i2 * NT;
      float a3 = 0.f;
      for (int j = 0; j < NT; j++) a3 += g1[j] * g2[j];
      red[t] = a3;
    }
  }
  __syncthreads();
  if (t == 0) {
    float s3 = 0.f;
    for (int p = 0; p < 256; p++) s3 += red[p] * red[256 + p];
    ws[W_RN] = sqrtf(s3) + 1.f;
  }
  __threadfence_block();
  __syncthreads();

  // ---- phase 7: amp + freq momentum updates ----
  const float lr = ws[W_LR];
  {
    float an = ws[W_AN];
    for (int idx = t; idx < NF * NT; idx += 1024) {
      float m = out[O_AGR + idx] * MOM + ws[W_AGR + idx] / an;
      out[O_AGR + idx] = m;
      out[O_ARE + idx] -= m * lr;
      m = out[O_AGI + idx] * MOM + ws[W_AGI + idx] / an;
      out[O_AGI + idx] = m;
      out[O_AIM + idx] -= m * lr;
    }
    if (t < NF) {
      float fn = ws[W_FN];
      float m = out[O_FGM + t] * MOM + ws[W_FG + t] / fn;
      out[O_FGM + t] = m;
      out[O_FRQ + t] -= m * lr;
    }
  }

  // ---- phase 8: rg = u^T(1536x16) @ g(16x512) via WMMA, fused update ----
  {
    const float rn = ws[W_RN];
    const float* u = ws + W_U;
    const float* g = ws + W_G;
    for (int tile = wv; tile < 3072; tile += 32) {   // 96 f-tiles x 32 j-tiles
      int fb = (tile >> 5) * 16;
      int jb = (tile & 31) * 16;
      v8f acc = {0.f, 0.f, 0.f, 0.f, 0.f, 0.f, 0.f, 0.f};
#pragma unroll
      for (int k = 0; k < NB; k += 4) {
        int kk = k + 2 * half;
        v2f a, bb;
        a.x = u[kk * NCAT + fb + q];           // A_{f,i} = u[i][f]
        a.y = u[(kk + 1) * NCAT + fb + q];
        bb.x = g[kk * NT + jb + q];            // B_{i,j} = g[i][j]
        bb.y = g[(kk + 1) * NT + jb + q];
        acc = wmma4(a, bb, acc);
      }
#pragma unroll
      for (int v = 0; v < 8; v++) {
        int idx = (fb + v + 8 * half) * NT + jb + q;
        float m = out[O_RPG + idx] * MOM + acc[v] / rn;
        out[O_RPG + idx] = m;
        out[O_RP + idx] -= m * lr;
      }
    }
  }
}

// ---------------- host ----------------
static inline void copy4(const float* src, float* dst, int n, hipStream_t s) {
  int n4 = n >> 2;
  int blocks = (n4 + 255) / 256;
  if (blocks > 4096) blocks = 4096;
  k_copy4<<<blocks, 256, 0, s>>>((const float4*)src, (float4*)dst, n4);
}

extern "C" void kernel_launch(void* const* d_in, const int* in_sizes, int n_in,
                              void* d_out, int out_size, void* d_ws, size_t ws_size,
                              hipStream_t stream) {
  (void)in_sizes; (void)n_in; (void)out_size; (void)ws_size;
  const int*   p_t   = (const int*)d_in[0];
  const int*   p_ds  = (const int*)d_in[1];
  const int*   ts    = (const int*)d_in[2];
  const float* x     = (const float*)d_in[3];
  const float* z     = (const float*)d_in[4];
  const float* mem   = (const float*)d_in[5];
  const int*   tmem  = (const int*)d_in[6];
  const float* feat  = (const float*)d_in[7];
  const float* are   = (const float*)d_in[8];
  const float* aim   = (const float*)d_in[9];
  const float* frq   = (const float*)d_in[10];
  const float* cfq   = (const float*)d_in[11];
  const float* rp    = (const float*)d_in[12];
  const float* fmean = (const float*)d_in[13];
  const float* bias  = (const float*)d_in[14];
  const float* agr   = (const float*)d_in[15];
  const float* agi   = (const float*)d_in[16];
  const float* fgr   = (const float*)d_in[17];
  const float* rpg   = (const float*)d_in[18];
  float* out = (float*)d_out;
  float* ws  = (float*)d_ws;

  // running means + lr
  k_init<<<1, 1024, 0, stream>>>(bias, z, fmean, x, p_ds, out, ws);
  // bulk state copy-out (dominates the byte budget; b128 vector path)
  copy4(mem,  out + O_MEM,  MEMSZ * NT, stream);
  copy4(feat, out + O_FEAT, MEMSZ * NFEAT, stream);
  copy4(are,  out + O_ARE,  NF * NT, stream);
  copy4(aim,  out + O_AIM,  NF * NT, stream);
  copy4(frq,  out + O_FRQ,  NF, stream);
  copy4(rp,   out + O_RP,   NCAT * NT, stream);
  copy4(agr,  out + O_AGR,  NF * NT, stream);
  copy4(agi,  out + O_AGI,  NF * NT, stream);
  copy4(fgr,  out + O_FGM,  NF, stream);
  copy4(rpg,  out + O_RPG,  NCAT * NT, stream);
  k_tmem<<<64, 256, 0, stream>>>(tmem, p_t, p_ds, out);
  k_fixrows<<<4, 256, 0, stream>>>(z, x, p_ds, out);

  // two sequential optimizer steps, fully fused (latency-bound work)
  for (int b = 0; b < 2; b++)
    k_batch<<<1, 1024, 0, stream>>>(ts, p_t, p_ds, tmem, mem, z, x, feat, cfq,
                                    out, ws, b);
}